// SimpleAttention_12146167513223
// MI455X (gfx1250) — compile-verified
//
#include <hip/hip_runtime.h>
#include <hip/hip_bf16.h>

// ---------------------------------------------------------------------------
// SimpleAttention on MI455X (gfx1250, wave32)
//   q,k,v = x@W{q,k,v}^T -> RoPE(q,k) -> causal flash attention -> @Wo^T
// All matmuls on V_WMMA_F32_16X16X4_F32 (native fp32 matrix path, exact vs
// the fp32 reference). Flash attention double-buffers K/V tiles in LDS via
// CDNA5 GLOBAL_LOAD_ASYNC_TO_LDS_B128 (ASYNCcnt-tracked, no VGPR round-trip).
// ---------------------------------------------------------------------------

typedef float v2f __attribute__((ext_vector_type(2)));
typedef float v8f __attribute__((ext_vector_type(8)));

#define B_SZ   2
#define S_LEN  2048
#define H_DIM  1024
#define NH     16
#define HD     64
#define SCALE  0.125f        // HD^-0.5

#define KV_STRIDE 72         // 16x64 tile rows padded: 288B, 16B-aligned, ~2-way banks
#define P_STRIDE  18         // 16x16 P tile rows padded: 8B-aligned, conflict-free

static __device__ __forceinline__ v8f wmma_f32(v2f a, v2f b, v8f c) {
  return __builtin_amdgcn_wmma_f32_16x16x4_f32(
      false, a, false, b, (short)0, c, false, false);
}

static __device__ __forceinline__ v8f zero8() {
  v8f z = {0.f, 0.f, 0.f, 0.f, 0.f, 0.f, 0.f, 0.f};
  return z;
}

// LDS aperture places the byte offset in addr[31:0] of a generic pointer.
static __device__ __forceinline__ uint32_t lds_off(const void* p) {
  return (uint32_t)(uintptr_t)p;
}

// CDNA5 async memory->LDS copy, 16B per lane, tracked by ASYNCcnt.
static __device__ __forceinline__ void async_load_b128(uint32_t lds_addr,
                                                       const void* gaddr) {
  asm volatile("global_load_async_to_lds_b128 %0, %1, off"
               :: "v"(lds_addr), "v"((uint64_t)(uintptr_t)gaddr) : "memory");
}

static __device__ __forceinline__ void wait_asynccnt0() {
  asm volatile("s_wait_asynccnt 0" ::: "memory");
}

// ---------------------------------------------------------------------------
// Kernel 1: fused QKV projection + RoPE.
// Each wave computes a 32(M) x 64(N) strip of one of q/k/v = x @ W^T
// (2 M-tiles share every B fragment: 6 loads per 8 WMMAs).
// Fragment layouts (wave32, V_WMMA_F32_16X16X4_F32):
//   A(16x4): element(lane,v): M = lane%16,  K = 2*(lane/16)+v
//   B(4x16): element(lane,v): N = lane%16,  K = 2*(lane/16)+v
//   C(16x16): element(lane,r): N = lane%16, M = r + 8*(lane/16)
// ---------------------------------------------------------------------------
__global__ void qkv_rope_kernel(const float* __restrict__ x,
                                const float* __restrict__ cosb,
                                const float* __restrict__ sinb,
                                const float* __restrict__ wq,
                                const float* __restrict__ wk,
                                const float* __restrict__ wv,
                                float* __restrict__ qout,
                                float* __restrict__ kout,
                                float* __restrict__ vout) {
  const int lane = threadIdx.x & 31;
  const int wib  = threadIdx.x >> 5;
  const int gw   = blockIdx.x * 8 + wib;           // 6144 waves total

  const int ngrp  = gw & 15;                       // 16 N-groups of 64 cols
  const int mtile = (gw >> 4) & 127;               // 128 M-tiles of 32 rows
  const int mat   = gw >> 11;                      // 0=q 1=k 2=v
  if (mat >= 3) return;

  const float* w   = (mat == 0) ? wq : (mat == 1) ? wk : wv;
  float*       out = (mat == 0) ? qout : (mat == 1) ? kout : vout;

  const int half = lane >> 4, l16 = lane & 15;
  const int m0 = mtile * 32;
  const int n0 = ngrp * 64;                        // == head * HD

  const float* ap0 = x + (size_t)(m0 + l16) * H_DIM + half * 2;
  const float* ap1 = ap0 + (size_t)16 * H_DIM;
  const float* bp0 = w + (size_t)(n0 + l16) * H_DIM + half * 2;

  v8f acc[8];                                      // [mt*4 + nt]
#pragma unroll
  for (int i = 0; i < 8; ++i) acc[i] = zero8();

  for (int k0 = 0; k0 < H_DIM; k0 += 4) {
    v2f a0 = *(const v2f*)(ap0 + k0);
    v2f a1 = *(const v2f*)(ap1 + k0);
#pragma unroll
    for (int nt = 0; nt < 4; ++nt) {
      v2f b = *(const v2f*)(bp0 + (size_t)(nt * 16) * H_DIM + k0);
      acc[nt]     = wmma_f32(a0, b, acc[nt]);
      acc[4 + nt] = wmma_f32(a1, b, acc[4 + nt]);
    }
  }

  const bool doRope = (mat < 2);
#pragma unroll
  for (int mt = 0; mt < 2; ++mt) {
#pragma unroll
    for (int nt = 0; nt < 4; ++nt) {
#pragma unroll
      for (int r = 0; r < 8; ++r) {
        const int m    = m0 + mt * 16 + r + 8 * half;  // flattened (b,s)
        const int bidx = m >> 11;
        const int s    = m & (S_LEN - 1);
        const int d    = nt * 16 + l16;
        float val = acc[mt * 4 + nt][r];
        if (doRope) {
          const float partner = __shfl_xor(val, 1, 32);   // pair d^1 in lane^1
          const float c  = cosb[s * (HD / 2) + (d >> 1)];
          const float sn = sinb[s * (HD / 2) + (d >> 1)];
          val = (d & 1) ? (partner * sn + val * c)
                        : (val * c - partner * sn);
        }
        out[(((size_t)bidx * NH + ngrp) * S_LEN + s) * HD + d] = val;
      }
    }
  }
}

// ---------------------------------------------------------------------------
// Kernel 2: causal flash attention.
// Block = 8 waves covering 8 consecutive 16-query tiles of one (batch, head).
// K/V 16x64 tiles are double-buffered in LDS via async memory->LDS copies:
//   iter i: [wait own ASYNCcnt; barrier] -> issue async prefetch of tile i+1
//           into buf (i+1)&1 -> compute on buf i&1.
// Overwrite hazard is fenced by the iteration-i barrier (buf (i+1)&1 was last
// READ in iteration i-1). All waves run the block-max trip count; compute is
// predicated wave-uniformly so EXEC stays all-ones for WMMA. P is transposed
// C->A layout through a per-wave padded LDS buffer with s_wait_dscnt.
// ---------------------------------------------------------------------------
__global__ void flash_kernel(const float* __restrict__ Q,
                             const float* __restrict__ K,
                             const float* __restrict__ V,
                             float* __restrict__ O /* [B,S,H] */) {
  __shared__ __align__(16) float kt[2][16 * KV_STRIDE];
  __shared__ __align__(16) float vt[2][16 * KV_STRIDE];
  __shared__ __align__(16) float pbuf[8 * 16 * P_STRIDE];

  const int lane = threadIdx.x & 31;
  const int wib  = threadIdx.x >> 5;
  float* pw = pbuf + wib * 16 * P_STRIDE;

  // 512 blocks: b * 16 heads * 16 query-blocks of 128
  const int bq = blockIdx.x & 15;
  const int h  = (blockIdx.x >> 4) & 15;
  const int b  = blockIdx.x >> 8;

  const int half = lane >> 4, l16 = lane & 15;
  const int q0   = bq * 128 + wib * 16;            // this wave's query tile
  const int nIter = 8 * bq + 8;                    // tiles up to block causal max

  const size_t headOff = ((size_t)b * NH + h) * S_LEN * HD;
  const float* qbase = Q + headOff;
  const float* kbase = K + headOff;
  const float* vbase = V + headOff;

  // cooperative tile coordinates: 256 threads x 16B = one 16x64 fp32 tile
  const int tr = threadIdx.x >> 4;                 // row 0..15
  const int tc = (threadIdx.x & 15) * 4;           // col (floats)
  const uint32_t klds0 = lds_off(&kt[0][tr * KV_STRIDE + tc]);
  const uint32_t klds1 = lds_off(&kt[1][tr * KV_STRIDE + tc]);
  const uint32_t vlds0 = lds_off(&vt[0][tr * KV_STRIDE + tc]);
  const uint32_t vlds1 = lds_off(&vt[1][tr * KV_STRIDE + tc]);
  const size_t gtile = (size_t)tr * HD + tc;

  // Preload Q A-fragments (K-dim 64 -> 16 WMMA K-steps)
  v2f qf[16];
  const float* qp = qbase + (size_t)(q0 + l16) * HD + half * 2;
#pragma unroll
  for (int s = 0; s < 16; ++s) qf[s] = *(const v2f*)(qp + 4 * s);

  v8f oacc[4];
#pragma unroll
  for (int i = 0; i < 4; ++i) oacc[i] = zero8();
  float mrow[8], lrow[8];
#pragma unroll
  for (int r = 0; r < 8; ++r) { mrow[r] = -3.0e38f; lrow[r] = 0.0f; }

  // prologue: async-stage tile 0 into buffer 0
  async_load_b128(klds0, kbase + gtile);
  async_load_b128(vlds0, vbase + gtile);

  for (int it = 0; it < nIter; ++it) {
    const int j0  = it * 16;
    const int cur = it & 1;

    wait_asynccnt0();                              // my tile-it copies landed
    __syncthreads();                               // all waves' copies visible;
                                                   // all reads of buf cur^1 done
    if (it + 1 < nIter) {                          // prefetch tile it+1
      const size_t gnext = (size_t)(it + 1) * 16 * HD + gtile;
      async_load_b128(cur ? klds0 : klds1, kbase + gnext);
      async_load_b128(cur ? vlds0 : vlds1, vbase + gnext);
    }

    if (j0 <= q0) {                                // wave-uniform predicate
      const float* ktc = kt[cur];
      const float* vtc = vt[cur];

      // ---- scores: S = Q * K^T (B fragments from LDS) ----
      v8f sacc = zero8();
#pragma unroll
      for (int s = 0; s < 16; ++s) {
        v2f bf = *(const v2f*)&ktc[l16 * KV_STRIDE + 4 * s + half * 2];
        sacc = wmma_f32(qf[s], bf, sacc);
      }

      const bool diag = (j0 == q0);
#pragma unroll
      for (int r = 0; r < 8; ++r) {
        float sc = sacc[r] * SCALE;
        const int qrow = q0 + r + 8 * half;
        const int kcol = j0 + l16;
        if (diag && (kcol > qrow)) sc = -3.0e38f;
        float rmax = sc;                           // reduce across 16-lane half
        rmax = fmaxf(rmax, __shfl_xor(rmax, 1, 32));
        rmax = fmaxf(rmax, __shfl_xor(rmax, 2, 32));
        rmax = fmaxf(rmax, __shfl_xor(rmax, 4, 32));
        rmax = fmaxf(rmax, __shfl_xor(rmax, 8, 32));
        const float mnew = fmaxf(mrow[r], rmax);
        const float p = __expf(sc - mnew);
        float rsum = p;
        rsum += __shfl_xor(rsum, 1, 32);
        rsum += __shfl_xor(rsum, 2, 32);
        rsum += __shfl_xor(rsum, 4, 32);
        rsum += __shfl_xor(rsum, 8, 32);
        const float alpha = __expf(mrow[r] - mnew);
        mrow[r] = mnew;
        lrow[r] = lrow[r] * alpha + rsum;
#pragma unroll
        for (int nt = 0; nt < 4; ++nt) oacc[nt][r] *= alpha;
        pw[(r + 8 * half) * P_STRIDE + l16] = p;   // P row-major [q][k]
      }
      // wave-private LDS store->load ordering (CDNA5 split counter)
      asm volatile("s_wait_dscnt 0" ::: "memory");

      // ---- O += P(16x16) * V(16x64), both operands from LDS ----
#pragma unroll
      for (int kk = 0; kk < 4; ++kk) {
        v2f af = *(const v2f*)&pw[l16 * P_STRIDE + kk * 4 + half * 2];
        const int vr = kk * 4 + half * 2;
#pragma unroll
        for (int nt = 0; nt < 4; ++nt) {
          v2f bf;
          bf.x = vtc[(vr)     * KV_STRIDE + nt * 16 + l16];
          bf.y = vtc[(vr + 1) * KV_STRIDE + nt * 16 + l16];
          oacc[nt] = wmma_f32(af, bf, oacc[nt]);
        }
      }
    }
  }

  // normalize, write attention output in [B,S,H] layout for the final GEMM
#pragma unroll
  for (int nt = 0; nt < 4; ++nt) {
#pragma unroll
    for (int r = 0; r < 8; ++r) {
      const int qrow = q0 + r + 8 * half;
      const int d = nt * 16 + l16;
      O[((size_t)b * S_LEN + qrow) * H_DIM + h * HD + d] = oacc[nt][r] / lrow[r];
    }
  }
}

// ---------------------------------------------------------------------------
// Kernel 3: output projection  out = attn @ Wo^T  (32x64 strip per wave)
// ---------------------------------------------------------------------------
__global__ void proj_kernel(const float* __restrict__ A,
                            const float* __restrict__ W,
                            float* __restrict__ Out) {
  const int lane = threadIdx.x & 31;
  const int wib  = threadIdx.x >> 5;
  const int gw   = blockIdx.x * 8 + wib;           // 2048 waves
  const int ngrp  = gw & 15;
  const int mtile = gw >> 4;                       // 0..127 (32-row tiles)
  if (mtile >= 128) return;

  const int half = lane >> 4, l16 = lane & 15;
  const int m0 = mtile * 32, n0 = ngrp * 64;
  const float* ap0 = A + (size_t)(m0 + l16) * H_DIM + half * 2;
  const float* ap1 = ap0 + (size_t)16 * H_DIM;
  const float* bp0 = W + (size_t)(n0 + l16) * H_DIM + half * 2;

  v8f acc[8];
#pragma unroll
  for (int i = 0; i < 8; ++i) acc[i] = zero8();

  for (int k0 = 0; k0 < H_DIM; k0 += 4) {
    v2f a0 = *(const v2f*)(ap0 + k0);
    v2f a1 = *(const v2f*)(ap1 + k0);
#pragma unroll
    for (int nt = 0; nt < 4; ++nt) {
      v2f b = *(const v2f*)(bp0 + (size_t)(nt * 16) * H_DIM + k0);
      acc[nt]     = wmma_f32(a0, b, acc[nt]);
      acc[4 + nt] = wmma_f32(a1, b, acc[4 + nt]);
    }
  }

#pragma unroll
  for (int mt = 0; mt < 2; ++mt) {
#pragma unroll
    for (int nt = 0; nt < 4; ++nt) {
#pragma unroll
      for (int r = 0; r < 8; ++r) {
        const int m = m0 + mt * 16 + r + 8 * half;
        const int n = n0 + nt * 16 + l16;
        Out[(size_t)m * H_DIM + n] = acc[mt * 4 + nt][r];
      }
    }
  }
}

// ---------------------------------------------------------------------------
extern "C" void kernel_launch(void* const* d_in, const int* in_sizes, int n_in,
                              void* d_out, int out_size, void* d_ws, size_t ws_size,
                              hipStream_t stream) {
  const float* x    = (const float*)d_in[0];
  const float* cosb = (const float*)d_in[1];
  const float* sinb = (const float*)d_in[2];
  const float* wq   = (const float*)d_in[3];
  const float* wk   = (const float*)d_in[4];
  const float* wv   = (const float*)d_in[5];
  const float* wo   = (const float*)d_in[6];

  float* ws = (float*)d_ws;
  const size_t QSZ = (size_t)B_SZ * NH * S_LEN * HD;   // 4.19M floats
  float* qb = ws;                 // RoPE'd Q  [B,NH,S,HD]
  float* kb = ws + QSZ;           // RoPE'd K
  float* vb = ws + 2 * QSZ;       // V
  float* ab = ws + 3 * QSZ;       // attention output [B,S,H]
  (void)ws_size; (void)in_sizes; (void)n_in; (void)out_size;

  // 3 mats * 128 M-tiles(32) * 16 N-groups = 6144 waves / 8 per block
  qkv_rope_kernel<<<768, 256, 0, stream>>>(x, cosb, sinb, wq, wk, wv, qb, kb, vb);
  // 2 * 16 heads * 16 query-blocks(128) = 512 blocks
  flash_kernel<<<512, 256, 0, stream>>>(qb, kb, vb, ab);
  // 128 * 16 = 2048 waves / 8 per block
  proj_kernel<<<256, 256, 0, stream>>>(ab, wo, (float*)d_out);
}